// DFTQNN_38242388803804
// MI455X (gfx1250) — compile-verified
//
#include <hip/hip_runtime.h>

#define DIM (1 << 24)      // 2^24 amplitudes
#define TILE 4096          // 8^4 elements per block tile
#define NBLK (DIM / TILE)  // 4096 blocks

typedef __attribute__((ext_vector_type(2))) float v2f;
typedef __attribute__((ext_vector_type(8))) float v8f;

// M = [[Gr, -Gi], [Gi, Gr]]  (16x16 real embedding of the 8x8 complex gate)
__device__ __forceinline__ float m_entry(const float* __restrict__ gr,
                                         const float* __restrict__ gi,
                                         int r, int c) {
  const int rr = r & 7, cc = c & 7;
  if (r < 8) {
    return (c < 8) ? gr[rr * 8 + cc] : -gi[rr * 8 + cc];
  } else {
    return (c < 8) ? gi[rr * 8 + cc] : gr[rr * 8 + cc];
  }
}

// Per-lane A-operand chunks for V_WMMA_F32_16X16X4_F32.
// A chunk kc is M[:, 4kc:4kc+4]; lane L<16 holds {M[L][4kc], M[L][4kc+1]},
// lane L>=16 holds {M[L-16][4kc+2], M[L-16][4kc+3]}.
__device__ __forceinline__ void make_a(const float* __restrict__ gr,
                                       const float* __restrict__ gi,
                                       int lane, v2f* a) {
  const int r = lane & 15;
  const int h2 = (lane >> 4) << 1;
#pragma unroll
  for (int kc = 0; kc < 4; ++kc) {
    const int c0 = 4 * kc + h2;
    a[kc].x = m_entry(gr, gi, r, c0);
    a[kc].y = m_entry(gr, gi, r, c0 + 1);
  }
}

// Apply the 16x16 real gate M along the axis with tile-stride S.
// Tile in LDS: lds[0..4095] = re plane, lds[4096..8191] = im plane.
// 512 fibers per axis; 8 waves x 16 columns x 4 groups cover all of them.
// D = sum_kc A_kc * B_kc : B chunk kc holds X rows [4kc,4kc+4) striped per ISA
// (v0 = row 4kc+2hi, v1 = row 4kc+2hi+1). Rows 0-7 of X = re, 8-15 = im.
// D rows 0-7 = new re (lanes 0-15), rows 8-15 = new im (lanes 16-31).
template <int S>
__device__ __forceinline__ void apply_axis(float* lds, const v2f* a, int lane,
                                           int wave) {
  float* lre = lds;
  float* lim = lds + TILE;
  const int col = lane & 15;
  const int hi = lane >> 4;
  const int h2 = hi << 1;
#pragma unroll
  for (int g = 0; g < 4; ++g) {
    const int f = (wave * 4 + g) * 16 + col;            // fiber id in [0,512)
    const int base = (f / S) * (8 * S) + (f % S);       // fiber base in tile
    v2f b0 = {lre[base + (h2 + 0) * S], lre[base + (h2 + 1) * S]};
    v2f b1 = {lre[base + (h2 + 4) * S], lre[base + (h2 + 5) * S]};
    v2f b2 = {lim[base + (h2 + 0) * S], lim[base + (h2 + 1) * S]};
    v2f b3 = {lim[base + (h2 + 4) * S], lim[base + (h2 + 5) * S]};
    v8f acc = {};
    acc = __builtin_amdgcn_wmma_f32_16x16x4_f32(false, a[0], false, b0,
                                                (short)0, acc, false, false);
    acc = __builtin_amdgcn_wmma_f32_16x16x4_f32(false, a[1], false, b1,
                                                (short)0, acc, false, false);
    acc = __builtin_amdgcn_wmma_f32_16x16x4_f32(false, a[2], false, b2,
                                                (short)0, acc, false, false);
    acc = __builtin_amdgcn_wmma_f32_16x16x4_f32(false, a[3], false, b3,
                                                (short)0, acc, false, false);
    float* outp = lds + hi * TILE;  // hi=0 -> re plane, hi=1 -> im plane
#pragma unroll
    for (int j = 0; j < 8; ++j) outp[base + j * S] = acc[j];
  }
}

// ---------------- normalization (deterministic two-stage reduction) ---------
__global__ __launch_bounds__(256) void k_reduce_partial(
    const float* __restrict__ f, float* __restrict__ partial) {
  __shared__ float sdata[256];
  const int tid = threadIdx.x;
  const float4* f4 = (const float4*)f + (size_t)blockIdx.x * 1024;
  float s = 0.f;
#pragma unroll
  for (int i = 0; i < 4; ++i) {
    float4 v = f4[tid + 256 * i];
    s += v.x * v.x + v.y * v.y + v.z * v.z + v.w * v.w;
  }
  sdata[tid] = s;
  __syncthreads();
  for (int off = 128; off > 0; off >>= 1) {
    if (tid < off) sdata[tid] += sdata[tid + off];
    __syncthreads();
  }
  if (tid == 0) partial[blockIdx.x] = sdata[0];
}

__global__ __launch_bounds__(256) void k_reduce_final(
    const float* __restrict__ partial, float* __restrict__ inv_out) {
  __shared__ float sdata[256];
  const int tid = threadIdx.x;
  float s = 0.f;
#pragma unroll
  for (int i = 0; i < 16; ++i) s += partial[tid + 256 * i];
  sdata[tid] = s;
  __syncthreads();
  for (int off = 128; off > 0; off >>= 1) {
    if (tid < off) sdata[tid] += sdata[tid + off];
    __syncthreads();
  }
  if (tid == 0) inv_out[0] = 1.0f / sdata[0];  // probs scale = 1/||f||^2
}

// ---------------- pass 1: axes with global strides 1,8,64,512 ---------------
__global__ __launch_bounds__(256) void k_inner(
    const float* __restrict__ feat, const float* __restrict__ gr,
    const float* __restrict__ gi, float* __restrict__ out_re,
    float* __restrict__ out_im) {
  __shared__ float lds[2 * TILE];
  const int tid = threadIdx.x;
  const int lane = tid & 31, wave = tid >> 5;
  const int gb4 = blockIdx.x * (TILE / 4);  // float4 base
  float4* l4 = (float4*)lds;
  const float4* f4 = (const float4*)feat;
  v2f a[4];
  make_a(gr, gi, lane, a);
  const float4 z4 = {0.f, 0.f, 0.f, 0.f};
#pragma unroll
  for (int i = 0; i < 4; ++i) {
    const int q = tid + 256 * i;  // float4 index in tile [0,1024)
    l4[q] = f4[gb4 + q];
    l4[1024 + q] = z4;
  }
  __syncthreads();
  apply_axis<1>(lds, a, lane, wave);
  __syncthreads();
  apply_axis<8>(lds, a, lane, wave);
  __syncthreads();
  apply_axis<64>(lds, a, lane, wave);
  __syncthreads();
  apply_axis<512>(lds, a, lane, wave);
  __syncthreads();
  float4* or4 = (float4*)out_re;
  float4* oi4 = (float4*)out_im;
#pragma unroll
  for (int i = 0; i < 4; ++i) {
    const int q = tid + 256 * i;
    or4[gb4 + q] = l4[q];
    oi4[gb4 + q] = l4[1024 + q];
  }
}

// ---------------- pass 2: axes with global strides 4096, 32768 --------------
// Tile = 64 contiguous elements x 64 outer combos: g = base + c + 4096*o,
// base = 64*m + 262144*n. Tile strides for the two axes: 64 and 512.
__global__ __launch_bounds__(256) void k_mid(float* __restrict__ st_re,
                                             float* __restrict__ st_im,
                                             const float* __restrict__ gr,
                                             const float* __restrict__ gi) {
  __shared__ float lds[2 * TILE];
  const int tid = threadIdx.x;
  const int lane = tid & 31, wave = tid >> 5;
  const int m = blockIdx.x & 63, n = blockIdx.x >> 6;
  const int base4 = (64 * m + 262144 * n) >> 2;  // float4 index of tile base
  float4* l4 = (float4*)lds;
  float4* re4 = (float4*)st_re;
  float4* im4 = (float4*)st_im;
  v2f a[4];
  make_a(gr, gi, lane, a);
#pragma unroll
  for (int i = 0; i < 4; ++i) {
    const int q = tid + 256 * i;        // float4 index in tile [0,1024)
    const int c4 = q & 15, o = q >> 4;  // 16 float4 per 64-float row
    const int g4 = base4 + 1024 * o + c4;  // 4096 floats = 1024 float4 stride
    l4[q] = re4[g4];
    l4[1024 + q] = im4[g4];
  }
  __syncthreads();
  apply_axis<64>(lds, a, lane, wave);
  __syncthreads();
  apply_axis<512>(lds, a, lane, wave);
  __syncthreads();
#pragma unroll
  for (int i = 0; i < 4; ++i) {
    const int q = tid + 256 * i;
    const int c4 = q & 15, o = q >> 4;
    const int g4 = base4 + 1024 * o + c4;
    re4[g4] = l4[q];
    im4[g4] = l4[1024 + q];
  }
}

// ---------------- pass 3: axes with strides 262144, 2097152 + |amp|^2 -------
// g = base + c + 262144*o, base = 64*m. Writes probs in place of re plane.
__global__ __launch_bounds__(256) void k_top(float* __restrict__ st_re_probs,
                                             const float* __restrict__ st_im,
                                             const float* __restrict__ gr,
                                             const float* __restrict__ gi,
                                             const float* __restrict__ invp) {
  __shared__ float lds[2 * TILE];
  const int tid = threadIdx.x;
  const int lane = tid & 31, wave = tid >> 5;
  const int base4 = (blockIdx.x * 64) >> 2;  // float4 index of tile base
  float4* l4 = (float4*)lds;
  float4* re4 = (float4*)st_re_probs;
  const float4* im4 = (const float4*)st_im;
  v2f a[4];
  make_a(gr, gi, lane, a);
#pragma unroll
  for (int i = 0; i < 4; ++i) {
    const int q = tid + 256 * i;
    const int c4 = q & 15, o = q >> 4;
    const int g4 = base4 + 65536 * o + c4;  // 262144 floats = 65536 float4
    l4[q] = re4[g4];
    l4[1024 + q] = im4[g4];
  }
  __syncthreads();
  apply_axis<64>(lds, a, lane, wave);
  __syncthreads();
  apply_axis<512>(lds, a, lane, wave);
  __syncthreads();
  const float inv = invp[0];
#pragma unroll
  for (int i = 0; i < 4; ++i) {
    const int q = tid + 256 * i;
    const int c4 = q & 15, o = q >> 4;
    const int g4 = base4 + 65536 * o + c4;
    const float4 r = l4[q], s = l4[1024 + q];
    float4 p;
    p.x = (r.x * r.x + s.x * s.x) * inv;
    p.y = (r.y * r.y + s.y * s.y) * inv;
    p.z = (r.z * r.z + s.z * s.z) * inv;
    p.w = (r.w * r.w + s.w * s.w) * inv;
    re4[g4] = p;
  }
}

extern "C" void kernel_launch(void* const* d_in, const int* in_sizes, int n_in,
                              void* d_out, int out_size, void* d_ws,
                              size_t ws_size, hipStream_t stream) {
  (void)in_sizes; (void)n_in; (void)out_size; (void)ws_size;
  const float* feat = (const float*)d_in[0];  // (DIM,) f32
  const float* gr = (const float*)d_in[1];    // (8,8) f32
  const float* gi = (const float*)d_in[2];    // (8,8) f32
  float* out = (float*)d_out;                 // DIM f32 probs; also re plane
  float* ws = (float*)d_ws;
  float* im_plane = ws;                  // DIM floats
  float* partial = ws + DIM;             // 4096 floats
  float* inv = partial + NBLK;           // 1 float

  k_reduce_partial<<<NBLK, 256, 0, stream>>>(feat, partial);
  k_reduce_final<<<1, 256, 0, stream>>>(partial, inv);
  k_inner<<<NBLK, 256, 0, stream>>>(feat, gr, gi, out, im_plane);
  k_mid<<<NBLK, 256, 0, stream>>>(out, im_plane, gr, gi);
  k_top<<<NBLK, 256, 0, stream>>>(out, im_plane, gr, gi, inv);
}